// RadianceField_35553739276602
// MI455X (gfx1250) — compile-verified
//
#include <hip/hip_runtime.h>
#include <math.h>

#define IDIM   128
#define SUSED  115          // S-1 samples actually contribute
#define SPAD   128          // padded per-ray stride in workspace (float4 units)

typedef float v2f __attribute__((ext_vector_type(2)));
typedef float v8f __attribute__((ext_vector_type(8)));

__device__ __forceinline__ float clamp0h(float v) { return fminf(fmaxf(v, 0.0f), 100000.0f); }

__device__ __forceinline__ float sh_pick(int j, float s0, float s1, float s2, float s3,
                                         float s4, float s5, float s6, float s7, float s8) {
  float r = s0;
  r = (j == 1) ? s1 : r;  r = (j == 2) ? s2 : r;  r = (j == 3) ? s3 : r;  r = (j == 4) ? s4 : r;
  r = (j == 5) ? s5 : r;  r = (j == 6) ? s6 : r;  r = (j == 7) ? s7 : r;  r = (j == 8) ? s8 : r;
  return r;
}

// ---------------- Phase 1: one block per ray, one thread per sample ----------------
__global__ __launch_bounds__(128)
void rf_phase1(const float* __restrict__ xg, const float* __restrict__ dg,
               const float* __restrict__ tmin, const float* __restrict__ tmax,
               const float* __restrict__ grid, const float* __restrict__ opac,
               float4* __restrict__ ws)
{
  const int ray  = blockIdx.x;
  const int s    = threadIdx.x;                 // 0..127
  const int se   = (s < SUSED) ? s : (SUSED - 1); // clamp so all lanes stay live (EXEC all-1 for WMMA)
  const int lane = threadIdx.x & 31;

  const float ox = xg[3 * ray], oy = xg[3 * ray + 1], oz = xg[3 * ray + 2];
  const float dx = dg[3 * ray], dy = dg[3 * ray + 1], dz = dg[3 * ray + 2];
  const float t0 = tmin[ray], t1 = tmax[ray];

  // SH bases (per ray)
  const float SH_C0 = 0.28209479177387814f, SH_C1 = 0.4886025119029199f;
  const float s0 = SH_C0;
  const float s1 = -SH_C1 * dy;
  const float s2 =  SH_C1 * dz;
  const float s3 = -SH_C1 * dx;
  const float s4 =  1.0925484305920792f * dx * dy;
  const float s5 = -1.0925484305920792f * dy * dz;
  const float s6 =  0.31539156525252005f * (2.0f * dz * dz - dx * dx - dy * dy);
  const float s7 = -1.0925484305920792f * dx * dz;
  const float s8 =  0.5462742152960396f * (dx * dx - dy * dy);

  const float t  = t0 + (t1 - t0) * (0.05f + (float)se * (1.0f / 128.0f));
  const float px = fminf(fmaxf(fmaf(t, dx, ox), 0.0f), 126.9999f);
  const float py = fminf(fmaxf(fmaf(t, dy, oy), 0.0f), 126.9999f);
  const float pz = fminf(fmaxf(fmaf(t, dz, oz), 0.0f), 126.9999f);
  const int ix = (int)floorf(px), iy = (int)floorf(py), iz = (int)floorf(pz);
  const float wx = px - (float)ix, wy = py - (float)iy, wz = pz - (float)iz;

  // Trilerp: accumulate the 27-channel coefficient vector + opacity.
  float c[28];
#pragma unroll
  for (int k = 0; k < 28; ++k) c[k] = 0.0f;   // c[27] stays 0 => K padding
  float op = 0.0f;

#pragma unroll
  for (int cn = 0; cn < 8; ++cn) {
    const int bx = (cn >> 2) & 1, by = (cn >> 1) & 1, bz = cn & 1;
    const float wc = (bx ? wx : 1.0f - wx) * (by ? wy : 1.0f - wy) * (bz ? wz : 1.0f - wz);
    const int vi = ((ix + bx) * IDIM + (iy + by)) * IDIM + (iz + bz);
    const float* __restrict__ g = grid + (size_t)vi * 27u;
#pragma unroll
    for (int k = 0; k < 27; ++k) c[k] = fmaf(wc, g[k], c[k]);
    op = fmaf(wc, opac[vi], op);
  }
  op = clamp0h(op);

  // SH projection: D[c][n] = 0.5 + sum_k shmat[k][c] * coeffs[n][k]   (K = 28, 7 tiles of 4)
  float cr, cg, cb;
#if __has_builtin(__builtin_amdgcn_wmma_f32_16x16x4_f32)
  {
    const int m  = lane & 15;            // A-matrix row held by this lane
    const int ks = (lane < 16) ? 0 : 2;  // which K pair this lane holds
    v2f af[7];
#pragma unroll
    for (int kt = 0; kt < 7; ++kt) {
      const int ka = kt * 4 + ks, kb = ka + 1;
      const int ca = ka / 9, ckb = kb / 9;
      const float va = sh_pick(ka - ca * 9, s0, s1, s2, s3, s4, s5, s6, s7, s8);
      const float vb = sh_pick(kb - ckb * 9, s0, s1, s2, s3, s4, s5, s6, s7, s8);
      af[kt].x = (ka < 27 && ca == m) ? va : 0.0f;
      af[kt].y = (kb < 27 && ckb == m) ? vb : 0.0f;
    }
    v8f d1 = {0.5f, 0.5f, 0.5f, 0.0f, 0.0f, 0.0f, 0.0f, 0.0f};  // +0.5 bias in rows 0..2
    v8f d2 = {0.5f, 0.5f, 0.5f, 0.0f, 0.0f, 0.0f, 0.0f, 0.0f};
#pragma unroll
    for (int kt = 0; kt < 7; ++kt) {
      const int k0 = kt * 4;
      // B tile for samples = wave lanes 0..15
      const float p2 = __shfl(c[k0 + 2], lane & 15, 32);
      const float p3 = __shfl(c[k0 + 3], lane & 15, 32);
      v2f b0; b0.x = (lane < 16) ? c[k0] : p2;  b0.y = (lane < 16) ? c[k0 + 1] : p3;
      d1 = __builtin_amdgcn_wmma_f32_16x16x4_f32(false, af[kt], false, b0, (short)0, d1, false, false);
      // B tile for samples = wave lanes 16..31
      const float q0 = __shfl(c[k0],     16 + (lane & 15), 32);
      const float q1 = __shfl(c[k0 + 1], 16 + (lane & 15), 32);
      v2f b1; b1.x = (lane < 16) ? q0 : c[k0 + 2];  b1.y = (lane < 16) ? q1 : c[k0 + 3];
      d2 = __builtin_amdgcn_wmma_f32_16x16x4_f32(false, af[kt], false, b1, (short)0, d2, false, false);
    }
    // D[c][n] lives at VGPR c, lane n (n = 0..15). Upper-half lanes fetch their column.
    const float e0 = __shfl(d2[0], lane & 15, 32);
    const float e1 = __shfl(d2[1], lane & 15, 32);
    const float e2 = __shfl(d2[2], lane & 15, 32);
    cr = (lane < 16) ? d1[0] : e0;
    cg = (lane < 16) ? d1[1] : e1;
    cb = (lane < 16) ? d1[2] : e2;
  }
#else
  #define DOT9(B) (c[B]*s0 + c[B+1]*s1 + c[B+2]*s2 + c[B+3]*s3 + c[B+4]*s4 + \
                   c[B+5]*s5 + c[B+6]*s6 + c[B+7]*s7 + c[B+8]*s8)
  cr = 0.5f + DOT9(0);
  cg = 0.5f + DOT9(9);
  cb = 0.5f + DOT9(18);
  #undef DOT9
#endif
  cr = clamp0h(cr); cg = clamp0h(cg); cb = clamp0h(cb);

  if (s < SUSED) ws[(size_t)ray * SPAD + s] = make_float4(cr, cg, cb, op);
}

// ---------------- Phase 2: one wave32 per ray — scan + composite ----------------
__global__ __launch_bounds__(256)
void rf_phase2(const float* __restrict__ tmin, const float* __restrict__ tmax,
               const float4* __restrict__ ws, float* __restrict__ out, int nray)
{
  const int wid  = (int)((blockIdx.x * blockDim.x + threadIdx.x) >> 5);
  const int lane = threadIdx.x & 31;
  if (wid >= nray) return;

  const float delta = (tmax[wid] - tmin[wid]) * (1.0f / 128.0f);
  float ar = 0.0f, ag = 0.0f, ab = 0.0f;
  float carry = 0.0f;

#pragma unroll
  for (int ch = 0; ch < 4; ++ch) {
    const int sidx = ch * 32 + lane;
    float4 v = make_float4(0.0f, 0.0f, 0.0f, 0.0f);
    if (sidx < SUSED) v = ws[(size_t)wid * SPAD + sidx];
    const float dts = -delta * v.w;          // 0 for padding lanes -> weight 0
    // inclusive wave scan of dts
    float sc = dts;
#pragma unroll
    for (int off = 1; off < 32; off <<= 1) {
      const float n = __shfl_up(sc, off, 32);
      if (lane >= off) sc += n;
    }
    const float excl = carry + (sc - dts);   // exclusive prefix (== reference cum)
    const float w = expf(excl) * (1.0f - expf(dts));
    ar = fmaf(w, v.x, ar);
    ag = fmaf(w, v.y, ag);
    ab = fmaf(w, v.z, ab);
    carry += __shfl(sc, 31, 32);
  }
  // cross-lane reduction
#pragma unroll
  for (int off = 16; off; off >>= 1) {
    ar += __shfl_down(ar, off, 32);
    ag += __shfl_down(ag, off, 32);
    ab += __shfl_down(ab, off, 32);
  }
  if (lane == 0) {
    out[3 * wid + 0] = ar;
    out[3 * wid + 1] = ag;
    out[3 * wid + 2] = ab;
  }
}

// ---------------- Fallback: monolithic one-thread-per-ray (no workspace) ----------------
__global__ __launch_bounds__(256)
void rf_mono(const float* __restrict__ xg, const float* __restrict__ dg,
             const float* __restrict__ tmin, const float* __restrict__ tmax,
             const float* __restrict__ grid, const float* __restrict__ opac,
             float* __restrict__ out, int nray)
{
  const int ray = blockIdx.x * blockDim.x + threadIdx.x;
  if (ray >= nray) return;

  const float ox = xg[3 * ray], oy = xg[3 * ray + 1], oz = xg[3 * ray + 2];
  const float dx = dg[3 * ray], dy = dg[3 * ray + 1], dz = dg[3 * ray + 2];
  const float t0 = tmin[ray], t1 = tmax[ray];

  const float SH_C0 = 0.28209479177387814f, SH_C1 = 0.4886025119029199f;
  float shv[9];
  shv[0] = SH_C0;            shv[1] = -SH_C1 * dy;      shv[2] = SH_C1 * dz;
  shv[3] = -SH_C1 * dx;      shv[4] = 1.0925484305920792f * dx * dy;
  shv[5] = -1.0925484305920792f * dy * dz;
  shv[6] = 0.31539156525252005f * (2.0f * dz * dz - dx * dx - dy * dy);
  shv[7] = -1.0925484305920792f * dx * dz;
  shv[8] = 0.5462742152960396f * (dx * dx - dy * dy);

  const float delta = (t1 - t0) * (1.0f / 128.0f);
  float cum = 0.0f, ar = 0.0f, ag = 0.0f, ab = 0.0f;

  for (int si = 0; si < SUSED; ++si) {
    const float t  = t0 + (t1 - t0) * (0.05f + (float)si * (1.0f / 128.0f));
    const float px = fminf(fmaxf(fmaf(t, dx, ox), 0.0f), 126.9999f);
    const float py = fminf(fmaxf(fmaf(t, dy, oy), 0.0f), 126.9999f);
    const float pz = fminf(fmaxf(fmaf(t, dz, oz), 0.0f), 126.9999f);
    const int ix = (int)floorf(px), iy = (int)floorf(py), iz = (int)floorf(pz);
    const float wx = px - (float)ix, wy = py - (float)iy, wz = pz - (float)iz;

    float rr = 0.0f, rg = 0.0f, rb = 0.0f, op = 0.0f;
#pragma unroll
    for (int cn = 0; cn < 8; ++cn) {
      const int bx = (cn >> 2) & 1, by = (cn >> 1) & 1, bz = cn & 1;
      const float wc = (bx ? wx : 1.0f - wx) * (by ? wy : 1.0f - wy) * (bz ? wz : 1.0f - wz);
      const int vi = ((ix + bx) * IDIM + (iy + by)) * IDIM + (iz + bz);
      const float* __restrict__ g = grid + (size_t)vi * 27u;
      float d0 = 0.0f, d1 = 0.0f, d2 = 0.0f;
#pragma unroll
      for (int j = 0; j < 9; ++j) {
        d0 = fmaf(g[j],      shv[j], d0);
        d1 = fmaf(g[9 + j],  shv[j], d1);
        d2 = fmaf(g[18 + j], shv[j], d2);
      }
      rr = fmaf(wc, d0, rr);
      rg = fmaf(wc, d1, rg);
      rb = fmaf(wc, d2, rb);
      op = fmaf(wc, opac[vi], op);
    }
    op = clamp0h(op);
    const float cr = clamp0h(rr + 0.5f);
    const float cg = clamp0h(rg + 0.5f);
    const float cb = clamp0h(rb + 0.5f);

    const float dts = -delta * op;
    const float w = expf(cum) * (1.0f - expf(dts));
    ar = fmaf(w, cr, ar);
    ag = fmaf(w, cg, ag);
    ab = fmaf(w, cb, ab);
    cum += dts;
  }
  out[3 * ray + 0] = ar;
  out[3 * ray + 1] = ag;
  out[3 * ray + 2] = ab;
}

extern "C" void kernel_launch(void* const* d_in, const int* in_sizes, int n_in,
                              void* d_out, int out_size, void* d_ws, size_t ws_size,
                              hipStream_t stream)
{
  const float* x    = (const float*)d_in[0];
  const float* d    = (const float*)d_in[1];
  const float* tmn  = (const float*)d_in[2];
  const float* tmx  = (const float*)d_in[3];
  const float* grid = (const float*)d_in[4];
  const float* opac = (const float*)d_in[5];
  float* out = (float*)d_out;
  const int R = in_sizes[2];                         // number of rays (tmin element count)
  const size_t need = (size_t)R * SPAD * sizeof(float4);

  if (d_ws != nullptr && ws_size >= need) {
    rf_phase1<<<R, 128, 0, stream>>>(x, d, tmn, tmx, grid, opac, (float4*)d_ws);
    const int blocks = (R * 32 + 255) / 256;
    rf_phase2<<<blocks, 256, 0, stream>>>(tmn, tmx, (const float4*)d_ws, out, R);
  } else {
    rf_mono<<<(R + 255) / 256, 256, 0, stream>>>(x, d, tmn, tmx, grid, opac, out, R);
  }
}